// CA_ProteinFeatures_3607772528732
// MI455X (gfx1250) — compile-verified
//
#include <hip/hip_runtime.h>
#include <math.h>

#define Bsz  2
#define Lsz  4096
#define TOPK 30
#define EIN  167
#define EINP 168
#define EOUT 128
#define NEDGE (Bsz*Lsz*TOPK)   /* 245760 */
#define ROWS 32                /* edges per fused block */

typedef __attribute__((ext_vector_type(2))) float v2f;
typedef __attribute__((ext_vector_type(8))) float v8f;

// ---------------- kernel 0: pad edge_W (128x167) -> W2 (128x168), col 167 = 0
__global__ void k_padW(const float* __restrict__ W, float* __restrict__ W2) {
    int t = blockIdx.x * blockDim.x + threadIdx.x;
    if (t >= EOUT * EINP) return;
    int n = t / EINP, k = t - n * EINP;
    W2[t] = (k < EIN) ? W[n * EIN + k] : 0.0f;
}

// ---------------- kernel 1: masked pairwise distance + top-30 (stable ties)
__global__ void __launch_bounds__(64) k_topk(const float* __restrict__ Ca,
                                             const float* __restrict__ mask,
                                             float* __restrict__ Dnb,
                                             int*   __restrict__ Eidx,
                                             float* __restrict__ EidxF) {
    __shared__ float CaS[Lsz * 3];
    __shared__ float dl[TOPK][64];
    __shared__ int   il[TOPK][64];
    const int tid = threadIdx.x;
    const int b   = blockIdx.x >> 6;                    // 64 blocks per batch
    const int i   = ((blockIdx.x & 63) << 6) + tid;
    for (int t = tid; t < Lsz * 3; t += 64) CaS[t] = Ca[b * Lsz * 3 + t];
    __syncthreads();
    const float xi = CaS[i*3+0], yi = CaS[i*3+1], zi = CaS[i*3+2];
    const float mi = mask[b * Lsz + i];
    // pass 1: row max of masked D
    float dmax = 0.0f;
    for (int j = 0; j < Lsz; ++j) {
        float dx = xi - CaS[j*3+0], dy = yi - CaS[j*3+1], dz = zi - CaS[j*3+2];
        float m2 = mi * mask[b * Lsz + j];
        float D  = m2 * sqrtf(dx*dx + dy*dy + dz*dz + 1e-6f);
        dmax = fmaxf(dmax, D);
    }
    for (int s = 0; s < TOPK; ++s) { dl[s][tid] = 3.4e38f; il[s][tid] = 0; }
    // pass 2: insertion top-k on D_adjust (ascending j => stable ties like top_k)
    for (int j = 0; j < Lsz; ++j) {
        float dx = xi - CaS[j*3+0], dy = yi - CaS[j*3+1], dz = zi - CaS[j*3+2];
        float m2 = mi * mask[b * Lsz + j];
        float D  = m2 * sqrtf(dx*dx + dy*dy + dz*dz + 1e-6f);
        float Da = D + (1.0f - m2) * dmax;
        if (Da < dl[TOPK-1][tid]) {
            int s = TOPK - 1;
            while (s > 0 && Da < dl[s-1][tid]) {
                dl[s][tid] = dl[s-1][tid]; il[s][tid] = il[s-1][tid]; --s;
            }
            dl[s][tid] = Da; il[s][tid] = j;
        }
    }
    const int base = (b * Lsz + i) * TOPK;
    for (int s = 0; s < TOPK; ++s) {
        Dnb [base + s] = dl[s][tid];
        Eidx[base + s] = il[s][tid];
        EidxF[base + s] = (float)il[s][tid];
    }
}

// ---------------- kernel 2: coarse orientation frames O[b][i][9]
__device__ inline void unitU(const float* __restrict__ Ca, int t, float u[3]) {
    float dx = Ca[(t+1)*3+0] - Ca[t*3+0];
    float dy = Ca[(t+1)*3+1] - Ca[t*3+1];
    float dz = Ca[(t+1)*3+2] - Ca[t*3+2];
    float n  = sqrtf(dx*dx + dy*dy + dz*dz);
    float m  = (n > 3.6f && n < 4.0f) ? 1.0f : 0.0f;
    float den = fmaxf(m * n, 1e-12f);
    u[0] = dx * m / den; u[1] = dy * m / den; u[2] = dz * m / den;
}

__global__ void k_orient(const float* __restrict__ Ca, float* __restrict__ O) {
    int t = blockIdx.x * blockDim.x + threadIdx.x;
    if (t >= Bsz * Lsz) return;
    int b = t / Lsz, i = t - b * Lsz;
    float o[9] = {0,0,0, 0,0,0, 0,0,0};
    if (i >= 1 && i <= Lsz - 3) {
        const float* C = Ca + b * Lsz * 3;
        float u2[3], u1[3];
        unitU(C, i - 1, u2);
        unitU(C, i, u1);
        float cx = u2[1]*u1[2] - u2[2]*u1[1];
        float cy = u2[2]*u1[0] - u2[0]*u1[2];
        float cz = u2[0]*u1[1] - u2[1]*u1[0];
        float cn = fmaxf(sqrtf(cx*cx + cy*cy + cz*cz), 1e-12f);
        float n2x = cx/cn, n2y = cy/cn, n2z = cz/cn;
        float ox = u2[0]-u1[0], oy = u2[1]-u1[1], oz = u2[2]-u1[2];
        float on = fmaxf(sqrtf(ox*ox + oy*oy + oz*oz), 1e-12f);
        float o1x = ox/on, o1y = oy/on, o1z = oz/on;
        float c2x = o1y*n2z - o1z*n2y;
        float c2y = o1z*n2x - o1x*n2z;
        float c2z = o1x*n2y - o1y*n2x;
        o[0]=o1x; o[1]=o1y; o[2]=o1z;
        o[3]=n2x; o[4]=n2y; o[5]=n2z;
        o[6]=c2x; o[7]=c2y; o[8]=c2z;
    }
    for (int q = 0; q < 9; ++q) O[t*9 + q] = o[q];
}

// ---------------- fused kernel 3: features -> WMMA GEMM -> LayerNorm
__device__ inline float pdist3(const float a[3], const float bb[3]) {
    float dx = a[0]-bb[0], dy = a[1]-bb[1], dz = a[2]-bb[2];
    return sqrtf(dx*dx + dy*dy + dz*dz + 1e-6f);
}
__device__ inline void writeRBF(float* dst, float d) {
#pragma unroll
    for (int m = 0; m < 16; ++m) {
        float mu = 2.0f + (20.0f/15.0f) * (float)m;
        float t  = (d - mu) * (1.0f/1.25f);
        dst[m] = expf(-t*t);
    }
}
__device__ inline float sgnf(float x) { return (x > 0.f) ? 1.f : ((x < 0.f) ? -1.f : 0.f); }
__device__ inline void loadCa3(const float* __restrict__ Ca, int base, int idx, int ok, float v[3]) {
    if (ok) { v[0]=Ca[base+idx*3+0]; v[1]=Ca[base+idx*3+1]; v[2]=Ca[base+idx*3+2]; }
    else    { v[0]=0.f; v[1]=0.f; v[2]=0.f; }
}

__global__ void __launch_bounds__(128) k_edges(
    const float* __restrict__ Ca, const int* __restrict__ ridx, const int* __restrict__ chl,
    const float* __restrict__ posW, const float* __restrict__ posB,
    const float* __restrict__ W2, const float* __restrict__ lng, const float* __restrict__ lnb,
    const float* __restrict__ Ows, const float* __restrict__ Dnb, const int* __restrict__ Eidx,
    float* __restrict__ Eout)
{
    __shared__ __align__(16) float A[ROWS][172];     // padded K stride (bank-friendly)
    __shared__ __align__(16) float Elds[ROWS][129];  // stride 129 -> conflict-free LN
    const int tid = threadIdx.x;
    const int r0  = blockIdx.x * ROWS;

    // ---- phase 1: build 167-feature row per edge (threads 0..31)
    if (tid < ROWS) {
        const int e   = r0 + tid;
        const int b   = e / (Lsz * TOPK);
        const int rem = e - b * (Lsz * TOPK);
        const int i   = rem / TOPK;
        const int j   = Eidx[e];
        float* Ar = A[tid];
        // positional (16)
        {
            int off = ridx[b*Lsz + i] - ridx[b*Lsz + j];
            int ec  = (chl[b*Lsz + i] == chl[b*Lsz + j]);
            int dcode = ec ? min(max(off + 32, 0), 64) : 65;
#pragma unroll
            for (int p = 0; p < 16; ++p) Ar[p] = posW[p*66 + dcode] + posB[p];
        }
        // Ca triplets (shifted arrays zero-padded at chain ends)
        const int cb = b * Lsz * 3;
        float ci0[3], ci1[3], ci2[3], cj0[3], cj1[3], cj2[3];
        loadCa3(Ca, cb, i,   1,           ci1);
        loadCa3(Ca, cb, i-1, (i > 0),     ci0);
        loadCa3(Ca, cb, i+1, (i < Lsz-1), ci2);
        loadCa3(Ca, cb, j,   1,           cj1);
        loadCa3(Ca, cb, j-1, (j > 0),     cj0);
        loadCa3(Ca, cb, j+1, (j < Lsz-1), cj2);
        // RBFs (9 x 16)
        writeRBF(Ar + 16,  Dnb[e]);
        writeRBF(Ar + 32,  pdist3(ci0, cj0));
        writeRBF(Ar + 48,  pdist3(ci2, cj2));
        writeRBF(Ar + 64,  pdist3(ci0, cj1));
        writeRBF(Ar + 80,  pdist3(ci0, cj2));
        writeRBF(Ar + 96,  pdist3(ci1, cj0));
        writeRBF(Ar + 112, pdist3(ci1, cj2));
        writeRBF(Ar + 128, pdist3(ci2, cj0));
        writeRBF(Ar + 144, pdist3(ci2, cj1));
        // orientation features: dU (3) + quaternion (4)
        const float* Oi = Ows + (b*Lsz + i) * 9;
        const float* Oj = Ows + (b*Lsz + j) * 9;
        float dxn[3] = { cj1[0]-ci1[0], cj1[1]-ci1[1], cj1[2]-ci1[2] };
        float du[3];
#pragma unroll
        for (int r = 0; r < 3; ++r)
            du[r] = Oi[3*r+0]*dxn[0] + Oi[3*r+1]*dxn[1] + Oi[3*r+2]*dxn[2];
        float dn = fmaxf(sqrtf(du[0]*du[0] + du[1]*du[1] + du[2]*du[2]), 1e-12f);
        Ar[160] = du[0]/dn; Ar[161] = du[1]/dn; Ar[162] = du[2]/dn;
        float R[3][3];
#pragma unroll
        for (int r = 0; r < 3; ++r)
#pragma unroll
            for (int c = 0; c < 3; ++c)
                R[r][c] = Oi[0*3+r]*Oj[0*3+c] + Oi[1*3+r]*Oj[1*3+c] + Oi[2*3+r]*Oj[2*3+c];
        float Rxx = R[0][0], Ryy = R[1][1], Rzz = R[2][2];
        float m0 = 0.5f * sqrtf(fabsf(1.f + Rxx - Ryy - Rzz));
        float m1 = 0.5f * sqrtf(fabsf(1.f - Rxx + Ryy - Rzz));
        float m2 = 0.5f * sqrtf(fabsf(1.f - Rxx - Ryy + Rzz));
        float qx = sgnf(R[2][1] - R[1][2]) * m0;
        float qy = sgnf(R[0][2] - R[2][0]) * m1;
        float qz = sgnf(R[1][0] - R[0][1]) * m2;
        float qw = sqrtf(fmaxf(1.f + Rxx + Ryy + Rzz, 0.f)) * 0.5f;
        float qn = fmaxf(sqrtf(qx*qx + qy*qy + qz*qz + qw*qw), 1e-12f);
        Ar[163] = qx/qn; Ar[164] = qy/qn; Ar[165] = qz/qn; Ar[166] = qw/qn;
        Ar[167] = 0.0f;   // K padding (W2 col 167 is also 0)
    }
    __syncthreads();

    // ---- phase 2: WMMA GEMM  D[16x16] += A(16x4) x B(4x16), 42 K-steps
    const int wave = tid >> 5, lane = tid & 31;
    const int h = lane >> 4, m16 = lane & 15;   // A/B K-layout: reg r, half h -> K = 2h + r
    const int mt = wave >> 1;                   // M tile (0..1): rows mt*16..mt*16+15
    const int nbase = (wave & 1) * 64;          // this wave's 64 output columns
    const int arow = mt * 16 + m16;
    v8f acc[4] = {v8f{0.f}, v8f{0.f}, v8f{0.f}, v8f{0.f}};
    for (int kt = 0; kt < 42; ++kt) {
        const int kb = kt * 4 + 2 * h;          // even -> 8B-aligned float2
        const float2 a2 = *reinterpret_cast<const float2*>(&A[arow][kb]);
        const v2f av = {a2.x, a2.y};
#pragma unroll
        for (int nt = 0; nt < 4; ++nt) {
            const int n = nbase + nt * 16 + m16;
            const float2 b2 = *reinterpret_cast<const float2*>(W2 + n * EINP + kb);
            const v2f bv = {b2.x, b2.y};
            acc[nt] = __builtin_amdgcn_wmma_f32_16x16x4_f32(
                false, av, false, bv, (short)0, acc[nt], false, false);
        }
    }
    // spill accumulators: VGPR v -> row M = v + 8h, lane -> col N = m16
#pragma unroll
    for (int nt = 0; nt < 4; ++nt)
#pragma unroll
        for (int v = 0; v < 8; ++v)
            Elds[mt*16 + v + 8*h][nbase + nt*16 + m16] = acc[nt][v];
    __syncthreads();

    // ---- phase 3: LayerNorm over 128 + writeout
    if (tid < ROWS) {
        float s = 0.f;
        for (int c = 0; c < 128; ++c) s += Elds[tid][c];
        const float mu = s * (1.0f/128.0f);
        float vs = 0.f;
        for (int c = 0; c < 128; ++c) { float d = Elds[tid][c] - mu; vs += d*d; }
        const float rstd = rsqrtf(vs * (1.0f/128.0f) + 1e-5f);
        float* out = Eout + (size_t)(r0 + tid) * 128;
        for (int c = 0; c < 128; ++c)
            out[c] = (Elds[tid][c] - mu) * rstd * lng[c] + lnb[c];
    }
}

extern "C" void kernel_launch(void* const* d_in, const int* in_sizes, int n_in,
                              void* d_out, int out_size, void* d_ws, size_t ws_size,
                              hipStream_t stream) {
    (void)in_sizes; (void)n_in; (void)out_size; (void)ws_size;
    const float* Ca    = (const float*)d_in[0];
    const float* mask  = (const float*)d_in[1];
    const int*   ridx  = (const int*)  d_in[2];
    const int*   chl   = (const int*)  d_in[3];
    const float* posW  = (const float*)d_in[4];
    const float* posB  = (const float*)d_in[5];
    const float* edgeW = (const float*)d_in[6];
    const float* lng   = (const float*)d_in[7];
    const float* lnb   = (const float*)d_in[8];

    float* ws    = (float*)d_ws;
    float* W2    = ws;                        // 128*168          = 21504 f
    float* Ows   = W2  + EOUT * EINP;         // 2*4096*9         = 73728 f
    float* Dnb   = Ows + Bsz * Lsz * 9;       // 245760 f
    int*   Eidx  = (int*)(Dnb + NEDGE);       // 245760 i
    float* Eout  = (float*)d_out;             // 245760*128 f
    float* EidxF = Eout + (size_t)NEDGE * EOUT;

    hipLaunchKernelGGL(k_padW,  dim3((EOUT*EINP + 255)/256), dim3(256), 0, stream, edgeW, W2);
    hipLaunchKernelGGL(k_topk,  dim3(128), dim3(64), 0, stream, Ca, mask, Dnb, Eidx, EidxF);
    hipLaunchKernelGGL(k_orient, dim3((Bsz*Lsz + 255)/256), dim3(256), 0, stream, Ca, Ows);
    hipLaunchKernelGGL(k_edges, dim3(NEDGE / ROWS), dim3(128), 0, stream,
                       Ca, ridx, chl, posW, posB, W2, lng, lnb, Ows, Dnb, Eidx, Eout);
}